// Sampler_9156870275517
// MI455X (gfx1250) — compile-verified
//
#include <hip/hip_runtime.h>
#include <math.h>
#include <stdint.h>

// ---------------------------------------------------------------------------
// Fused sampler for MI455X (gfx1250, wave32).
//   grid = B rows, block = 1024 threads (32 waves) per row.
//   Sort-free top-p/top-k via 256-bin histogram select; histogram prefix
//   scans run on the matrix pipe (V_WMMA_F32_16X16X4_F32).
//   b128 vector memory ops; NT cache hints protect L2-resident ws;
//   8-way privatized LDS histograms kill hot-bin atomic serialization.
// ---------------------------------------------------------------------------

constexpr int Bn = 128;
constexpr int Vn = 128000;
constexpr int V4 = Vn / 4;
constexpr int Ln = 200;
constexpr int NBINS = 256;
constexpr int BDIM = 1024;
constexpr int NWAVE = BDIM / 32;
constexpr int NSLICE = 8;            // histogram privatization (4 waves/slice)
constexpr float BIN_SCALE = 16.0f;   // 256 bins over d = (max - x) in [0, 16)

typedef __attribute__((ext_vector_type(2))) float v2f;
typedef __attribute__((ext_vector_type(4))) float v4f;
typedef __attribute__((ext_vector_type(8))) float v8f;

__device__ __forceinline__ float wred_max(float v) {
#pragma unroll
  for (int o = 16; o > 0; o >>= 1) v = fmaxf(v, __shfl_xor(v, o, 32));
  return v;
}
__device__ __forceinline__ float wred_sum(float v) {
#pragma unroll
  for (int o = 16; o > 0; o >>= 1) v += __shfl_xor(v, o, 32);
  return v;
}
// Uniform broadcast from a known lane: v_readlane_b32 -> SGPR (no LDS).
__device__ __forceinline__ float readlane_f(float v, int srclane) {
  return __int_as_float(__builtin_amdgcn_readlane(__float_as_int(v), srclane));
}

// Deterministic per-(row, index) Gumbel(0,1) noise.
__device__ __forceinline__ float gumbel_hash(uint32_t row, uint32_t i) {
  uint32_t h = i * 0x9E3779B1u + row * 0x85EBCA77u + 0x165667B1u;
  h ^= h >> 16; h *= 0x85EBCA6Bu; h ^= h >> 13; h *= 0xC2B2AE35u; h ^= h >> 16;
  float u = ((h >> 8) + 0.5f) * (1.0f / 16777216.0f);   // (0,1)
  return -__logf(-__logf(u));
}

// 256-element inclusive prefix sum on one wave32 using V_WMMA_F32_16X16X4_F32.
//   View src as 16x16 row-major M:
//     rowscan = M x U          (U upper-tri ones; K=16 via 4 chained K=4 WMMAs)
//     offs    = Lstrict x RS   (RS[k][*] = rowsum[k], broadcast via readlane)
//     dst     = rowscan + offs
__device__ __forceinline__ void scan256_wmma(const float* src, float* dst,
                                             int lane) {
  const int half = lane >> 4;
  const int l = lane & 15;

  v8f acc = {};
#pragma unroll
  for (int q = 0; q < 4; ++q) {
    const int k0 = 4 * q + 2 * half;
    v2f a, b;
    a[0] = src[l * 16 + k0];
    a[1] = src[l * 16 + k0 + 1];
    b[0] = (k0 <= l) ? 1.0f : 0.0f;        // U[k][c] = (k <= c), c = l
    b[1] = (k0 + 1 <= l) ? 1.0f : 0.0f;
    acc = __builtin_amdgcn_wmma_f32_16x16x4_f32(false, a, false, b,
                                                (short)0, acc, false, false);
  }

  // rowsum[r] = rowscan(r, 15): lane 15 holds M=0..7, lane 31 holds M=8..15.
  float rs[16];
#pragma unroll
  for (int j = 0; j < 8; ++j) {
    rs[j] = readlane_f(acc[j], 15);
    rs[j + 8] = readlane_f(acc[j], 31);
  }

  v8f acc2 = {};
#pragma unroll
  for (int q = 0; q < 4; ++q) {
    const int k0 = 4 * q + 2 * half;
    v2f a, b;
    a[0] = (k0 < l) ? 1.0f : 0.0f;         // Lstrict[r][k] = (k < r), r = l
    a[1] = (k0 + 1 < l) ? 1.0f : 0.0f;
    b[0] = half ? rs[4 * q + 2] : rs[4 * q];
    b[1] = half ? rs[4 * q + 3] : rs[4 * q + 1];
    acc2 = __builtin_amdgcn_wmma_f32_16x16x4_f32(false, a, false, b,
                                                 (short)0, acc2, false, false);
  }
#pragma unroll
  for (int j = 0; j < 8; ++j) dst[(j + 8 * half) * 16 + l] = acc[j] + acc2[j];
}

__global__ __launch_bounds__(BDIM) void sampler_fused(
    const float* __restrict__ logits, const float* __restrict__ pres,
    const float* __restrict__ freq, const float* __restrict__ temps,
    const float* __restrict__ topps, const int* __restrict__ toks,
    const int* __restrict__ topks, float* __restrict__ out,
    float* __restrict__ ws) {
  // 128 KB packed u8 occurrence counts (L=200 < 256, no overflow),
  // 8-way privatized histograms (+16 KB), scan buffers. ~148 KB total,
  // well inside CDNA5's 320 KB WGP LDS.
  __shared__ uint32_t occ[V4];
  __shared__ uint32_t hcntW[NSLICE][NBINS];
  __shared__ float hexpW[NSLICE][NBINS];
  __shared__ float hexp[NBINS];
  __shared__ float scanC[NBINS];
  __shared__ float scanE[NBINS];
  __shared__ float redf[NWAVE];
  __shared__ int redi[NWAVE];
  __shared__ float s_max, s_Z, s_Zp;
  __shared__ int s_bk, s_bp, s_bLim, s_nt;

  const int row = blockIdx.x;
  const int tid = threadIdx.x;
  const int wid = tid >> 5;
  const int lane = tid & 31;
  const int slice = wid >> 2;          // 4 waves share one histogram slice

  for (int i = tid; i < V4; i += BDIM) occ[i] = 0u;
  for (int i = tid; i < NSLICE * NBINS; i += BDIM) {
    (&hcntW[0][0])[i] = 0u;
    (&hexpW[0][0])[i] = 0.0f;
  }
  if (tid == 0) { s_bk = NBINS - 1; s_bp = NBINS; }
  __syncthreads();

  // Penalty scatter: LDS ds_add_u32 atomics into packed byte counters.
  if (tid < Ln) {
    int t = toks[row * Ln + tid];
    atomicAdd(&occ[t >> 2], 1u << ((t & 3) << 3));
  }
  __syncthreads();

  const float pr = pres[row];
  const float fq = freq[row];
  const float invT = 1.0f / temps[row];
  const float tp = topps[row];
  const int k = topks[row];

  float* __restrict__ xrow = ws + (size_t)row * Vn;
  const v4f* __restrict__ lrow4 =
      reinterpret_cast<const v4f*>(logits + (size_t)row * Vn);
  v4f* __restrict__ xrow4 = reinterpret_cast<v4f*>(xrow);

  // ---- Pass 1 (HBM-bound, b128 NT loads): penalties + temp -> ws; max ------
  float vmax = -INFINITY;
  for (int i4 = tid; i4 < V4; i4 += BDIM) {
    if (i4 + 2 * BDIM < V4) __builtin_prefetch(&lrow4[i4 + 2 * BDIM], 0, 1);
    v4f lg = __builtin_nontemporal_load(&lrow4[i4]);  // read-once stream
    uint32_t w = occ[i4];  // 4 byte-counters for vocab ids 4*i4 .. 4*i4+3
    v4f x;
#pragma unroll
    for (int c = 0; c < 4; ++c) {
      uint32_t cc = (w >> (8 * c)) & 0xffu;
      x[c] = (lg[c] - (float)cc * fq - (cc ? pr : 0.0f)) * invT;
    }
    xrow4[i4] = x;   // RT policy: keep resident in L2 for passes 2-4
    vmax = fmaxf(vmax, fmaxf(fmaxf(x[0], x[1]), fmaxf(x[2], x[3])));
  }
  vmax = wred_max(vmax);
  if (lane == 0) redf[wid] = vmax;
  __syncthreads();
  if (wid == 0) {
    float v = (lane < NWAVE) ? redf[lane] : -INFINITY;
    v = wred_max(v);
    if (lane == 0) s_max = v;
  }
  __syncthreads();
  const float m = s_max;

  // ---- Pass 2 (L2-resident, b128): full Z + privatized histograms ----------
  float z = 0.0f;
  for (int i4 = tid; i4 < V4; i4 += BDIM) {
    v4f x = xrow4[i4];
#pragma unroll
    for (int c = 0; c < 4; ++c) {
      float d = m - x[c];
      float e = __expf(-d);
      z += e;
      if (d < 16.0f) {  // beyond 16: e < 2e-7, can never be the cutoff
        int b = (int)(d * BIN_SCALE);
        atomicAdd(&hcntW[slice][b], 1u);
        atomicAdd(&hexpW[slice][b], e);
      }
    }
  }
  z = wred_sum(z);
  if (lane == 0) redf[wid] = z;
  __syncthreads();
  if (wid == 0) {
    float v = (lane < NWAVE) ? redf[lane] : 0.0f;
    v = wred_sum(v);
    if (lane == 0) s_Z = v;
  }
  __syncthreads();

  // Reduce the 8 slices into the scan inputs.
  if (tid < NBINS) {
    uint32_t cs = 0u;
    float es = 0.0f;
#pragma unroll
    for (int s = 0; s < NSLICE; ++s) {
      cs += hcntW[s][tid];
      es += hexpW[s][tid];
    }
    scanC[tid] = (float)cs;
    hexp[tid] = es;
  }
  __syncthreads();

  // ---- Prefix scans of the 256 bins on the matrix pipe (wave 0) ------------
  if (wid == 0) {
    scan256_wmma(scanC, scanC, lane);   // cumulative counts
    scan256_wmma(hexp, scanE, lane);    // cumulative exp-mass
  }
  __syncthreads();

  // ---- Cutoff bin: top-k (cumcount >= k) and top-p (excl mass > p*Z) -------
  if (tid < NBINS) {
    if (scanC[tid] >= (float)k) atomicMin(&s_bk, tid);
    float excl = scanE[tid] - hexp[tid];
    if (excl > tp * s_Z) atomicMin(&s_bp, tid);
  }
  __syncthreads();
  if (tid == 0) {
    int bl = min(s_bk, s_bp - 1);
    s_bLim = bl < 0 ? 0 : bl;
  }
  __syncthreads();
  // keep iff bin(d) <= bLim  <=>  d*16 < bLim+1
  const float dLimScaled = (float)(s_bLim + 1);

  // ---- Pass 3 (L2-resident, b128): masked Z' + fused Gumbel-argmax ---------
  // argmax(logprob + g) = argmax over kept of (x + g): no log Z' needed here.
  float zp = 0.0f;
  float best = -INFINITY;
  int besti = 0;
  for (int i4 = tid; i4 < V4; i4 += BDIM) {
    v4f x = xrow4[i4];
#pragma unroll
    for (int c = 0; c < 4; ++c) {
      float d = m - x[c];
      if (d * BIN_SCALE < dLimScaled) {
        zp += __expf(-d);
        int i = 4 * i4 + c;
        float v = x[c] + gumbel_hash((uint32_t)row, (uint32_t)i);
        if (v > best) { best = v; besti = i; }
      }
    }
  }
  zp = wred_sum(zp);
#pragma unroll
  for (int o = 16; o > 0; o >>= 1) {
    float ov = __shfl_xor(best, o, 32);
    int oi = __shfl_xor(besti, o, 32);
    if (ov > best || (ov == best && oi < besti)) { best = ov; besti = oi; }
  }
  if (lane == 0) { redf[wid] = zp; scanE[wid] = best; redi[wid] = besti; }
  __syncthreads();
  if (wid == 0) {
    float v = (lane < NWAVE) ? redf[lane] : 0.0f;
    v = wred_sum(v);
    float bv = (lane < NWAVE) ? scanE[lane] : -INFINITY;
    int bi = (lane < NWAVE) ? redi[lane] : 0;
#pragma unroll
    for (int o = 16; o > 0; o >>= 1) {
      float ov = __shfl_xor(bv, o, 32);
      int oi = __shfl_xor(bi, o, 32);
      if (ov > bv || (ov == bv && oi < bi)) { bv = ov; bi = oi; }
    }
    if (lane == 0) { s_Zp = v; s_nt = bi; }
  }
  __syncthreads();
  const float invZp = 1.0f / s_Zp;
  const float logZp = __logf(s_Zp);

  // ---- Pass 4 (HBM store-bound, b128 NT stores): probs + logprobs ----------
  v4f* __restrict__ probs4 = reinterpret_cast<v4f*>(out + (size_t)row * Vn);
  v4f* __restrict__ lps4 =
      reinterpret_cast<v4f*>(out + (size_t)Bn * Vn + (size_t)row * Vn);
  for (int i4 = tid; i4 < V4; i4 += BDIM) {
    v4f x = xrow4[i4];
    v4f p, lp;
#pragma unroll
    for (int c = 0; c < 4; ++c) {
      float d = m - x[c];
      bool keep = (d * BIN_SCALE) < dLimScaled;
      float e = __expf(-d);
      p[c] = keep ? e * invZp : 0.0f;
      lp[c] = keep ? (-d - logZp) : -INFINITY;
    }
    // Write-once outputs: NT so they don't evict the L2-resident ws rows.
    __builtin_nontemporal_store(p, &probs4[i4]);
    __builtin_nontemporal_store(lp, &lps4[i4]);
  }
  if (tid == 0) out[2 * (size_t)Bn * Vn + row] = (float)s_nt;
}

extern "C" void kernel_launch(void* const* d_in, const int* in_sizes, int n_in,
                              void* d_out, int out_size, void* d_ws,
                              size_t ws_size, hipStream_t stream) {
  (void)in_sizes; (void)n_in; (void)out_size; (void)ws_size;
  const float* logits = (const float*)d_in[0];
  const float* pres   = (const float*)d_in[1];
  const float* freq   = (const float*)d_in[2];
  const float* temps  = (const float*)d_in[3];
  const float* topps  = (const float*)d_in[4];
  const int*   toks   = (const int*)d_in[5];
  const int*   topks  = (const int*)d_in[6];
  sampler_fused<<<Bn, BDIM, 0, stream>>>(logits, pres, freq, temps, topps,
                                         toks, topks, (float*)d_out,
                                         (float*)d_ws);
}